// IcoDiNeConv_8641474199777
// MI455X (gfx1250) — compile-verified
//
#include <hip/hip_runtime.h>
#include <hip/hip_bf16.h>
#include <cstddef>

// ---------------------------------------------------------------------------
// IcoDiNeConv: out[b,o,n] = sum_{c,k} x[b,c,neigh[n,k]] * W[o,c,k] + bias[o]
// B=8, C_IN=C_OUT=64, N_VERT=40962, K=7  ->  GEMM per (b, vertex-tile):
//   M = output channels (64), N = vertices (tile of 16), Kdim = 448 (j=k*64+c)
// fp32 path via V_WMMA_F32_16X16X4_F32 (preserves reference fp32 numerics).
// Gather staged through LDS with GLOBAL_LOAD_ASYNC_TO_LDS_B128 (ASYNCcnt).
// ---------------------------------------------------------------------------

typedef float v2f __attribute__((ext_vector_type(2)));
typedef float v8f __attribute__((ext_vector_type(8)));
typedef int   v4i __attribute__((vector_size(16)));

#define NVERT   40962
#define NBATCH  8
#define CIN     64
#define COUT    64
#define KNB     7
#define KD      448          // CIN*KNB, GEMM reduction length
#define STRIDE  452          // padded LDS row stride (452 % 64 == 4 -> no bank conflicts)
#define VTILES  ((NVERT + 15) / 16)   // 2561

#define GLOBAL_AS __attribute__((address_space(1)))
#define SHARED_AS __attribute__((address_space(3)))

// Async global -> LDS copy of 16 bytes, tracked by ASYNCcnt (no VGPR return).
__device__ __forceinline__ void async_copy_b128(const float* gsrc, float* ldst) {
  __builtin_amdgcn_global_load_async_to_lds_b128(
      (GLOBAL_AS v4i*)(gsrc), (SHARED_AS v4i*)(ldst), 0, 0);
}

__device__ __forceinline__ void wait_asynccnt0() {
  asm volatile("s_wait_asynccnt 0x0" ::: "memory");
}

// ---------------------------------------------------------------------------
// Pre-pass: transpose x (B, C, N) -> xT (B, N, C) so that the per-neighbor
// gather in the main kernel is a contiguous 256B row read.
// ---------------------------------------------------------------------------
__global__ __launch_bounds__(256) void transpose_x_kernel(
    const float* __restrict__ x, float* __restrict__ xT) {
  __shared__ float tile[64][65];
  const int blk = blockIdx.x;          // b * 641 + vblk
  const int b  = blk / 641;
  const int v0 = (blk % 641) * 64;
  const int tid = threadIdx.x;
  #pragma unroll
  for (int i = 0; i < 16; ++i) {
    int idx = tid + i * 256;           // 0..4095
    int c = idx >> 6, v = idx & 63;
    float val = 0.0f;
    if (v0 + v < NVERT)
      val = x[((size_t)b * CIN + c) * NVERT + v0 + v];   // coalesced in v
    tile[c][v] = val;
  }
  __syncthreads();
  #pragma unroll
  for (int i = 0; i < 16; ++i) {
    int idx = tid + i * 256;
    int v = idx >> 6, c = idx & 63;
    if (v0 + v < NVERT)
      xT[((size_t)b * NVERT + v0 + v) * CIN + c] = tile[c][v];  // coalesced in c
  }
}

// ---------------------------------------------------------------------------
// Main kernel: 128 threads = 4 waves. Wave w computes the 16x16 output tile
// (channels [16w,16w+16) x 16 vertices) with 112 f32 WMMAs over Kdim=448.
// LDS: aw[64][452]  = W permuted to j=k*64+c layout   (shared, filled once)
//      bt[16][452]  = gathered x rows for this vertex tile (refilled per tile
//                     via async global->LDS b128 copies)
// ---------------------------------------------------------------------------
__global__ __launch_bounds__(128, 1) void iconv_wmma_kernel(
    const float* __restrict__ xT,
    const long long* __restrict__ nidx,
    const float* __restrict__ W,
    const float* __restrict__ bias,
    float* __restrict__ out) {
  extern __shared__ float smem[];
  float* aw = smem;                    // 64 * 452 floats
  float* bt = smem + COUT * STRIDE;    // 16 * 452 floats

  const int tid  = threadIdx.x;
  const int lane = tid & 31;
  const int wid  = tid >> 5;           // 0..3
  const int ln   = lane & 15;
  const int half = lane >> 4;          // 0/1 : K-sub-pair & M-offset select
  const int otile = wid << 4;

  // Fill permuted weight matrix: aw[o][j] = W[o][c*7+k], j = k*64 + c.
  // Thread pair per channel; no divisions in the loop.
  {
    int o = tid >> 1;
    int j0 = (tid & 1) * 224;
    const float* wrow = W + o * KD;
    float* arow = aw + o * STRIDE;
    for (int j = j0; j < j0 + 224; ++j)
      arow[j] = wrow[(j & 63) * KNB + (j >> 6)];
  }

  // Per-lane bias fragment: D element (vgpr v, half h) is channel otile+v+8h.
  v8f biasfrag;
  {
    int ob = otile + (half << 3);
    #pragma unroll
    for (int v = 0; v < 8; ++v) biasfrag[v] = bias[ob + v];
  }

  // Loop-invariant gather assignment: thread tid < 112 owns row (vi, k) of
  // the gathered tile and copies one contiguous 64-float xT row per tile.
  const bool gactive = tid < 112;
  const int  gvi = tid / KNB;          // computed once, not in the hot loop
  const int  gk  = tid - gvi * KNB;
  float* gdst = bt + gvi * STRIDE + (gk << 6);

  __syncthreads();

  const int totalTiles = NBATCH * VTILES;
  for (int t = blockIdx.x; t < totalTiles; t += gridDim.x) {
    const int b  = t / VTILES;
    const int vt = (t - b * VTILES) << 4;

    // Async gather: each active lane streams its 256B row into LDS.
    if (gactive) {
      int vsrc = vt + gvi; if (vsrc >= NVERT) vsrc = NVERT - 1;  // clamp tail
      long long nb = nidx[(size_t)vsrc * KNB + gk];
      const float* gsrc = xT + ((size_t)b * NVERT + nb) * CIN;
      #pragma unroll
      for (int f = 0; f < 16; ++f)
        async_copy_b128(gsrc + (f << 2), gdst + (f << 2));
    }
    wait_asynccnt0();
    __syncthreads();

    // 112 WMMAs, two accumulators to break the D->C dependency chain.
    v8f acc0 = biasfrag;
    v8f acc1 = {0.f, 0.f, 0.f, 0.f, 0.f, 0.f, 0.f, 0.f};
    const float* ap = aw + (otile + ln) * STRIDE + (half << 1);
    const float* bp = bt + ln * STRIDE + (half << 1);
    #pragma unroll 4
    for (int kk = 0; kk < 112; kk += 2) {
      v2f a0 = *(const v2f*)(ap + (kk << 2));
      v2f b0 = *(const v2f*)(bp + (kk << 2));
      v2f a1 = *(const v2f*)(ap + ((kk + 1) << 2));
      v2f b1 = *(const v2f*)(bp + ((kk + 1) << 2));
      acc0 = __builtin_amdgcn_wmma_f32_16x16x4_f32(
          false, a0, false, b0, (short)0, acc0, false, false);
      acc1 = __builtin_amdgcn_wmma_f32_16x16x4_f32(
          false, a1, false, b1, (short)0, acc1, false, false);
    }
    v8f acc = acc0 + acc1;

    // Store: lane ln = vertex vt+ln; vgpr v -> channel otile + v + 8*half.
    // Per store, lanes 0-15 and 16-31 each write 64B contiguous runs.
    int vg = vt + ln;
    if (vg < NVERT) {
      float* op = out + ((size_t)((b << 6) + otile + (half << 3))) * NVERT + vg;
      #pragma unroll
      for (int v = 0; v < 8; ++v) op[(size_t)v * NVERT] = acc[v];
    }
    __syncthreads();   // protect bt before next tile's refill
  }
}

extern "C" void kernel_launch(void* const* d_in, const int* in_sizes, int n_in,
                              void* d_out, int out_size, void* d_ws, size_t ws_size,
                              hipStream_t stream) {
  const float*     x    = (const float*)d_in[0];
  const long long* nidx = (const long long*)d_in[1];
  const float*     W    = (const float*)d_in[2];
  const float*     bias = (const float*)d_in[3];
  float*           out  = (float*)d_out;
  float*           xT   = (float*)d_ws;   // 8*40962*64 floats = 83.9 MB

  (void)in_sizes; (void)n_in; (void)out_size; (void)ws_size;

  // Pre-pass transpose: 8 batches * 641 column tiles.
  transpose_x_kernel<<<NBATCH * 641, 256, 0, stream>>>(x, xT);

  // Main GEMM: dynamic LDS = (64 + 16) * 452 floats = 144,640 B (<=320KB/WGP).
  size_t shmem = (size_t)(COUT + 16) * STRIDE * sizeof(float);
  (void)hipFuncSetAttribute((const void*)iconv_wmma_kernel,
                            hipFuncAttributeMaxDynamicSharedMemorySize,
                            (int)shmem);
  iconv_wmma_kernel<<<2048, 128, shmem, stream>>>(xT, nidx, W, bias, out);
}